// Spatial_64106681860403
// MI455X (gfx1250) — compile-verified
//
#include <hip/hip_runtime.h>
#include <hip/hip_bf16.h>

typedef float v2f __attribute__((ext_vector_type(2)));
typedef float v4f __attribute__((ext_vector_type(4)));
typedef float v8f __attribute__((ext_vector_type(8)));

#define BATCH            512
#define HW               50176          // 224*224
#define FLOATS_PER_BATCH (HW * 3)       // 150528
#define F4_PER_BATCH     (FLOATS_PER_BATCH / 4)   // 37632
#define GROUPS_PER_BATCH (FLOATS_PER_BATCH / 12)  // 12544 groups of 4 pixels
#define BLOCKS_X         7              // 12544 = 7 * 7 * 256
#define ITERS            7              // groups per thread
#define GROUP_STRIDE     (BLOCKS_X * 256)         // 1792 groups per sweep
#define EPSV             0.001f

#if __has_builtin(__builtin_amdgcn_wmma_f32_16x16x4_f32)
#define USE_WMMA 1
#else
#define USE_WMMA 0
#endif

// ---------------------------------------------------------------------------
// Zero the pooled accumulator (512 x 3 floats in workspace).
// ---------------------------------------------------------------------------
__global__ void init_ws(float* __restrict__ p, int n) {
    int i = blockIdx.x * blockDim.x + threadIdx.x;
    if (i < n) p[i] = 0.0f;
}

// ---------------------------------------------------------------------------
// Main streaming kernel: per-pixel sigmoid attention + weighted pooling.
// Grid (7, 512), 256 threads. Each thread: 7 groups x 3 b128 NT loads
// (28 pixels, 336 bytes) before one wave shfl reduction -> LDS cross-wave
// reduction -> 3 f32 global atomics. ~0.9 VALU/byte => HBM-bound.
// ---------------------------------------------------------------------------
__global__ __launch_bounds__(256)
void attn_pool(const float* __restrict__ in,
               const float* __restrict__ w1,
               const float* __restrict__ b1,
               float* __restrict__ pooled) {
    const int b    = blockIdx.y;
    const int gid0 = blockIdx.x * 256 + threadIdx.x;         // 0..1791

    const float W0 = w1[0], W1 = w1[1], W2 = w1[2], B1 = b1[0];
    const v4f* base4 = (const v4f*)in + (size_t)b * F4_PER_BATCH;

    float s0 = 0.0f, s1 = 0.0f, s2 = 0.0f;
#pragma unroll
    for (int it = 0; it < ITERS; ++it) {
        const v4f* in4 = base4 + (size_t)(gid0 + it * GROUP_STRIDE) * 3;
        v4f v0 = __builtin_nontemporal_load(in4 + 0);
        v4f v1 = __builtin_nontemporal_load(in4 + 1);
        v4f v2 = __builtin_nontemporal_load(in4 + 2);

        float px[4][3] = {
            {v0.x, v0.y, v0.z},
            {v0.w, v1.x, v1.y},
            {v1.z, v1.w, v2.x},
            {v2.y, v2.z, v2.w}};
#pragma unroll
        for (int p = 0; p < 4; ++p) {
            float z = px[p][0] * W0 + px[p][1] * W1 + px[p][2] * W2 + B1;
            // sigmoid(z) = rcp(1 + exp2(-z * log2(e)))
            float att = __builtin_amdgcn_rcpf(
                1.0f + __builtin_amdgcn_exp2f(z * -1.44269504088896340736f));
            s0 += att * px[p][0];
            s1 += att * px[p][1];
            s2 += att * px[p][2];
        }
    }

#pragma unroll
    for (int off = 16; off >= 1; off >>= 1) {
        s0 += __shfl_down(s0, off, 32);
        s1 += __shfl_down(s1, off, 32);
        s2 += __shfl_down(s2, off, 32);
    }

    __shared__ float wsum[8][3];
    const int lane = threadIdx.x & 31;
    const int wave = threadIdx.x >> 5;
    if (lane == 0) { wsum[wave][0] = s0; wsum[wave][1] = s1; wsum[wave][2] = s2; }
    __syncthreads();
    if (threadIdx.x < 3) {
        float t = 0.0f;
#pragma unroll
        for (int w = 0; w < 8; ++w) t += wsum[w][threadIdx.x];
        atomicAdd(&pooled[b * 3 + threadIdx.x], t);
    }
}

// ---------------------------------------------------------------------------
// Epilogue: x = pooled @ w2 + b2 ; LayerNorm(3) ; ReLU ; out = y @ w3 + b3.
// One 16-batch tile per wave with V_WMMA_F32_16X16X4_F32 (K = 3 padded to 4,
// N = 3 of 16). LN is a 3-wide row reduction done on the D-layout with
// 16-wide shuffles (DPP row_share); a tiny per-wave LDS transpose converts
// D-layout -> A-layout for the second WMMA.
// ---------------------------------------------------------------------------
__global__ __launch_bounds__(256)
void epilogue(const float* __restrict__ pooled,
              const float* __restrict__ w2, const float* __restrict__ b2,
              const float* __restrict__ gamma, const float* __restrict__ beta,
              const float* __restrict__ w3, const float* __restrict__ b3,
              float* __restrict__ out) {
#if USE_WMMA
    __shared__ float lds[8][16][4];          // per-wave 16 rows x K-pad(4)
    const int lane = threadIdx.x & 31;
    const int wave = threadIdx.x >> 5;
    const int base = (blockIdx.x * 8 + wave) * 16;   // first batch row of tile
    const int half = lane >> 4;                      // 0: lanes 0-15, 1: 16-31
    const int l16  = lane & 15;
    const int n    = l16;                            // B/C/D column index
    const int k0   = half ? 2 : 0;                   // A/B K slots per half
    const int k1   = half ? 3 : 1;

    // zero the LDS tile (K=3 pad column must be 0)
    for (int i = threadIdx.x; i < 8 * 16 * 4; i += 256)
        ((float*)lds)[i] = 0.0f;

    // A1: pooled tile, 16x4 f32 A-layout (lane<16: K0,K1 ; lane>=16: K2,K3)
    v2f a1;
    a1.x = pooled[(base + l16) * 3 + k0];
    a1.y = (k1 < 3) ? pooled[(base + l16) * 3 + k1] : 0.0f;
    // B1: w2 (C=3 x R=3) zero-padded to 4x16 B-layout
    v2f b1m;
    b1m.x = (n < 3 && k0 < 3) ? w2[k0 * 3 + n] : 0.0f;
    b1m.y = (n < 3 && k1 < 3) ? w2[k1 * 3 + n] : 0.0f;
    // C1: bias b2 broadcast down rows
    const float c1i = (n < 3) ? b2[n] : 0.0f;
    v8f c1;
#pragma unroll
    for (int j = 0; j < 8; ++j) c1[j] = c1i;

    v8f x1 = __builtin_amdgcn_wmma_f32_16x16x4_f32(false, a1, false, b1m,
                                                   (short)0, c1, false, false);

    const float g  = (n < 3) ? gamma[n] : 0.0f;
    const float bt = (n < 3) ? beta[n]  : 0.0f;
    __syncthreads();   // LDS zeros visible before LN writes

    // LayerNorm over the 3 valid columns of each row, then ReLU.
    // D-layout: VGPR j holds row j (lanes 0-15) / row j+8 (lanes 16-31).
#pragma unroll
    for (int j = 0; j < 8; ++j) {
        float v  = x1[j];
        float e0 = __shfl(v, 0, 16);
        float e1 = __shfl(v, 1, 16);
        float e2 = __shfl(v, 2, 16);
        float mean = (e0 + e1 + e2) * (1.0f / 3.0f);
        float var  = (e0 * e0 + e1 * e1 + e2 * e2) * (1.0f / 3.0f) - mean * mean;
        float y = (v - mean) * rsqrtf(var + EPSV) * g + bt;
        y = fmaxf(y, 0.0f);
        if (n < 3) lds[wave][j + half * 8][n] = y;
    }
    __syncthreads();

    // A2: y tile re-read in A-layout (column 3 is zero from the memset)
    v2f a2;
    a2.x = lds[wave][l16][k0];
    a2.y = lds[wave][l16][k1];
    // B2: w3 (R=3 x C=3) zero-padded
    v2f b2m;
    b2m.x = (n < 3 && k0 < 3) ? w3[k0 * 3 + n] : 0.0f;
    b2m.y = (n < 3 && k1 < 3) ? w3[k1 * 3 + n] : 0.0f;
    const float c2i = (n < 3) ? b3[n] : 0.0f;
    v8f c2;
#pragma unroll
    for (int j = 0; j < 8; ++j) c2[j] = c2i;

    v8f d = __builtin_amdgcn_wmma_f32_16x16x4_f32(false, a2, false, b2m,
                                                  (short)0, c2, false, false);
    if (n < 3) {
#pragma unroll
        for (int j = 0; j < 8; ++j)
            out[(base + j + half * 8) * 3 + n] = d[j];
    }
#else
    // Scalar fallback (never expected on gfx1250; kept for compile safety).
    int t = blockIdx.x * blockDim.x + threadIdx.x;
    if (t >= BATCH) return;
    float p0 = pooled[t * 3 + 0], p1 = pooled[t * 3 + 1], p2 = pooled[t * 3 + 2];
    float x[3], y[3];
#pragma unroll
    for (int r = 0; r < 3; ++r)
        x[r] = p0 * w2[0 * 3 + r] + p1 * w2[1 * 3 + r] + p2 * w2[2 * 3 + r] + b2[r];
    float mean = (x[0] + x[1] + x[2]) * (1.0f / 3.0f);
    float var  = (x[0] * x[0] + x[1] * x[1] + x[2] * x[2]) * (1.0f / 3.0f) - mean * mean;
    float rinv = rsqrtf(var + EPSV);
#pragma unroll
    for (int r = 0; r < 3; ++r)
        y[r] = fmaxf((x[r] - mean) * rinv * gamma[r] + beta[r], 0.0f);
#pragma unroll
    for (int c = 0; c < 3; ++c)
        out[t * 3 + c] = y[0] * w3[0 * 3 + c] + y[1] * w3[1 * 3 + c] + y[2] * w3[2 * 3 + c] + b3[c];
#endif
}

// ---------------------------------------------------------------------------
extern "C" void kernel_launch(void* const* d_in, const int* in_sizes, int n_in,
                              void* d_out, int out_size, void* d_ws, size_t ws_size,
                              hipStream_t stream) {
    const float* inputs = (const float*)d_in[0];
    const float* w1     = (const float*)d_in[1];
    const float* b1     = (const float*)d_in[2];
    const float* w2     = (const float*)d_in[3];
    const float* b2     = (const float*)d_in[4];
    const float* gamma  = (const float*)d_in[5];
    const float* beta   = (const float*)d_in[6];
    const float* w3     = (const float*)d_in[7];
    const float* b3     = (const float*)d_in[8];
    float* out    = (float*)d_out;
    float* pooled = (float*)d_ws;            // BATCH*3 floats of scratch

    init_ws<<<dim3((BATCH * 3 + 255) / 256), dim3(256), 0, stream>>>(pooled, BATCH * 3);

    dim3 grid(BLOCKS_X, BATCH);              // (7, 512), 7 groups per thread
    attn_pool<<<grid, dim3(256), 0, stream>>>(inputs, w1, b1, pooled);

    epilogue<<<dim3(BATCH / (16 * 8)), dim3(256), 0, stream>>>(  // 4 blocks x 8 waves
        pooled, w2, b2, gamma, beta, w3, b3, out);
}